// feed_forward_10144712753520
// MI455X (gfx1250) — compile-verified
//
#include <hip/hip_runtime.h>
#include <math.h>

typedef __attribute__((ext_vector_type(2))) float v2f;
typedef __attribute__((ext_vector_type(8))) float v8f;

#define FD 64
#define LDSW 66   // padded row stride (floats) to avoid 64-bank conflicts

// ---------------- zero workspace ----------------
__global__ void zero_ws_kernel(float* __restrict__ ws, long long n) {
    long long i = (long long)blockIdx.x * blockDim.x + threadIdx.x;
    long long stride = (long long)gridDim.x * blockDim.x;
    for (; i < n; i += stride) ws[i] = 0.0f;
}

// ---------------- edge scatter: wave per edge ----------------
// lane L handles channels 2L, 2L+1  (32 lanes x 2 = 64 channels, coalesced 256B)
__global__ void scatter_kernel(const float* __restrict__ x,
                               const long long* __restrict__ ei,
                               float* __restrict__ summed,
                               float* __restrict__ cnt,
                               int E) {
    int lane   = threadIdx.x & 31;
    int wave   = blockIdx.x * (blockDim.x >> 5) + (threadIdx.x >> 5);
    int nwaves = gridDim.x * (blockDim.x >> 5);
    for (int e = wave; e < E; e += nwaves) {
        long long s = ei[e];                 // edge_index[0][e] = src
        long long d = ei[(long long)E + e];  // edge_index[1][e] = dst
        const float2 v = *(const float2*)(x + s * FD + lane * 2);
        float* ds = summed + d * FD + lane * 2;
        unsafeAtomicAdd(ds,     v.x);        // global_atomic_add_f32 (hits L2)
        unsafeAtomicAdd(ds + 1, v.y);
        if (lane == 0) unsafeAtomicAdd(cnt + d, 1.0f);
    }
}

// ---------------- fused GEMM (f32 WMMA) + global-LN stats pass ----------------
// out[n][j] = sum_k mean_nb[n][k]*Wl[j][k] + b_l[j] + sum_k x[n][k]*Wr[j][k]
// One wave per 16-node tile; 8 waves / block; W_l, W_r staged in LDS.
__global__ void gemm_stats_kernel(const float* __restrict__ x,
                                  const float* __restrict__ summed,
                                  const float* __restrict__ cnt,
                                  const float* __restrict__ Wl,
                                  const float* __restrict__ bl,
                                  const float* __restrict__ Wr,
                                  float* __restrict__ out,
                                  float* __restrict__ stats,
                                  int N) {
    __shared__ float sW[2][FD * LDSW];
    __shared__ float red0[256];
    __shared__ float red1[256];

    // stage both weight matrices into LDS (row-padded)
    for (int i = threadIdx.x; i < FD * FD; i += blockDim.x) {
        int r = i >> 6, c = i & 63;
        sW[0][r * LDSW + c] = Wl[i];
        sW[1][r * LDSW + c] = Wr[i];
    }
    __syncthreads();

    const int lane  = threadIdx.x & 31;
    const int wave  = threadIdx.x >> 5;
    const int tile  = blockIdx.x * 8 + wave;   // 16 nodes per tile
    const int row   = lane & 15;               // A-row / B-col / D-col within tile
    const int khalf = lane >> 4;               // 0: K={k0,k0+1}, 1: K={k0+2,k0+3}

    float lsum = 0.0f, lsq = 0.0f;

    if (tile * 16 < N) {                       // wave-uniform guard (EXEC all-1 inside)
        const int n0 = tile * 16;
        const int n  = n0 + row;
        const float inv = 1.0f / fmaxf(cnt[n], 1.0f);

        // Hoist A fragments for the whole K range (16 chunks of K=4).
        v2f aM[16], aX[16];
        const float* mrow = summed + (long long)n * FD;
        const float* xrow = x      + (long long)n * FD;
#pragma unroll
        for (int kc = 0; kc < 16; ++kc) {
            const int kb = kc * 4 + khalf * 2;
            const float2 m  = *(const float2*)(mrow + kb);
            const float2 xv = *(const float2*)(xrow + kb);
            v2f am; am[0] = m.x * inv; am[1] = m.y * inv;  aM[kc] = am;
            v2f ax; ax[0] = xv.x;      ax[1] = xv.y;       aX[kc] = ax;
        }

#pragma unroll
        for (int jt = 0; jt < 4; ++jt) {       // 4 output column tiles of 16
            const int j = jt * 16 + row;
            const float bias = bl[j];
            v8f acc;
#pragma unroll
            for (int v = 0; v < 8; ++v) acc[v] = bias;   // b_l broadcast per column

#pragma unroll
            for (int kc = 0; kc < 16; ++kc) {
                const int kb = kc * 4 + khalf * 2;
                // B[k][j] = W[j][k]  -> lane reads float2 at W-row j, cols kb..kb+1
                const float2 b0 = *(const float2*)(&sW[0][j * LDSW + kb]);
                v2f vb0; vb0[0] = b0.x; vb0[1] = b0.y;
                acc = __builtin_amdgcn_wmma_f32_16x16x4_f32(
                    false, aM[kc], false, vb0, (short)0, acc, false, false);
                const float2 b1 = *(const float2*)(&sW[1][j * LDSW + kb]);
                v2f vb1; vb1[0] = b1.x; vb1[1] = b1.y;
                acc = __builtin_amdgcn_wmma_f32_16x16x4_f32(
                    false, aX[kc], false, vb1, (short)0, acc, false, false);
            }

            // D layout: VGPR v, lanes 0-15 -> M=v, lanes 16-31 -> M=v+8; col = row
#pragma unroll
            for (int v = 0; v < 8; ++v) {
                const int m = v + khalf * 8;
                const float val = acc[v];
                out[(long long)(n0 + m) * FD + j] = val;
                lsum += val;
                lsq  += val * val;
            }
        }
    }

    // block reduction of Sum and SumSq, then 2 atomics per block
    red0[threadIdx.x] = lsum;
    red1[threadIdx.x] = lsq;
    __syncthreads();
    for (int s = 128; s > 0; s >>= 1) {
        if ((int)threadIdx.x < s) {
            red0[threadIdx.x] += red0[threadIdx.x + s];
            red1[threadIdx.x] += red1[threadIdx.x + s];
        }
        __syncthreads();
    }
    if (threadIdx.x == 0) {
        unsafeAtomicAdd(&stats[0], red0[0]);
        unsafeAtomicAdd(&stats[1], red1[0]);
    }
}

// ---------------- normalize in place ----------------
__global__ void finalize_kernel(float* __restrict__ out,
                                const float* __restrict__ stats,
                                const float* __restrict__ lnw,
                                const float* __restrict__ lnb,
                                long long total) {
    const float inv_n = 1.0f / (float)total;
    const float mu    = stats[0] * inv_n;
    const float ex2   = stats[1] * inv_n;
    const float rstd  = rsqrtf(ex2 - mu * mu + 1e-5f);
    long long i = (long long)blockIdx.x * blockDim.x + threadIdx.x;
    const long long stride = (long long)gridDim.x * blockDim.x;
    for (; i < total; i += stride) {
        const int d = (int)(i & 63);
        out[i] = (out[i] - mu) * rstd * lnw[d] + lnb[d];
    }
}

extern "C" void kernel_launch(void* const* d_in, const int* in_sizes, int n_in,
                              void* d_out, int out_size, void* d_ws, size_t ws_size,
                              hipStream_t stream) {
    const float*      x   = (const float*)d_in[0];
    const long long*  ei  = (const long long*)d_in[1];
    const float*      Wl  = (const float*)d_in[2];
    const float*      bl  = (const float*)d_in[3];
    const float*      Wr  = (const float*)d_in[4];
    const float*      lnw = (const float*)d_in[5];
    const float*      lnb = (const float*)d_in[6];

    const int N = in_sizes[0] / FD;
    const int E = in_sizes[1] / 2;   // int64 element count = 2*E

    float* ws     = (float*)d_ws;
    float* summed = ws;                          // N*64 floats
    float* cnt    = ws + (long long)N * FD;      // N floats
    float* stats  = cnt + N;                     // 2 floats
    float* out    = (float*)d_out;

    const long long zn = (long long)N * FD + N + 2;
    zero_ws_kernel<<<1024, 256, 0, stream>>>(ws, zn);

    scatter_kernel<<<4096, 256, 0, stream>>>(x, ei, summed, cnt, E);

    const int tiles  = (N + 15) / 16;
    const int blocks = (tiles + 7) / 8;
    gemm_stats_kernel<<<blocks, 256, 0, stream>>>(x, summed, cnt, Wl, bl, Wr,
                                                  out, stats, N);

    finalize_kernel<<<2048, 256, 0, stream>>>(out, stats, lnw, lnb,
                                              (long long)N * FD);
}